// GetCommons_56023553409391
// MI455X (gfx1250) — compile-verified
//
#include <hip/hip_runtime.h>
#include <hip/hip_bf16.h>

typedef float v2f __attribute__((ext_vector_type(2)));
typedef float v8f __attribute__((ext_vector_type(8)));
typedef unsigned int u32x4 __attribute__((ext_vector_type(4)));
typedef int i32x4 __attribute__((ext_vector_type(4)));
typedef int i32x8 __attribute__((ext_vector_type(8)));

#define NDIM 4096
#define ROW_T 256
#define CHUNK 512
#define PADST 516   // 512 + 2 (mid-row, col>=256) + 2 (row end) -> conflict-free b64
#define K_INL 1024

// 2^53 fixed-point scale for deterministic integer summation of probabilities
#define FPSCALE 9007199254740992.0

#if defined(__has_builtin)
#  if __has_builtin(__builtin_amdgcn_tensor_load_to_lds) && \
      __has_builtin(__builtin_amdgcn_s_wait_tensorcnt)
#    define HAVE_TDM 1
#  endif
#endif
#ifndef HAVE_TDM
#  define HAVE_TDM 0
#endif

// ---------------------------------------------------------------------------
// Kernel 1: per-row radix-select threshold t*, kept-sum (-> 1/s), zeroed count
// ---------------------------------------------------------------------------
__global__ __launch_bounds__(ROW_T) void k_threshold(
    const float* __restrict__ scores, unsigned* __restrict__ thr,
    float* __restrict__ inv_s, int* __restrict__ inlier)
{
    __shared__ float srow[NDIM];                 // 16 KB
    __shared__ unsigned hcnt[256];
    __shared__ unsigned long long hsum[256];
    __shared__ unsigned s_prefix;
    __shared__ unsigned long long s_S;
    __shared__ unsigned s_tstar;
    __shared__ unsigned rc[ROW_T];
    __shared__ unsigned long long rs[ROW_T];

    const int r = blockIdx.x;                    // row id in [0, B*N)
    const int t = threadIdx.x;
    const float* src = scores + (size_t)r * NDIM;

    for (int i = t; i < NDIM; i += ROW_T) srow[i] = src[i];
    if (t == 0) { s_prefix = 0u; s_S = 0ull; s_tstar = 0u; }
    __syncthreads();

    const unsigned long long TH9 = (unsigned long long)(0.9 * FPSCALE);

    for (int p = 0; p < 4; ++p) {
        const int shift = 24 - 8 * p;
        hcnt[t] = 0u; hsum[t] = 0ull;
        __syncthreads();
        const unsigned prefix = s_prefix;
        for (int i = t; i < NDIM; i += ROW_T) {
            const float v = srow[i];
            const unsigned key = __float_as_uint(v);
            const bool match = (p == 0) ? true : ((key >> (shift + 8)) == prefix);
            if (match) {
                const unsigned d = (key >> shift) & 0xFFu;
                const unsigned long long q =
                    (unsigned long long)((double)v * FPSCALE);
                atomicAdd(&hcnt[d], 1u);
                atomicAdd(&hsum[d], q);          // integer: deterministic
            }
        }
        __syncthreads();
        if (t == 0) {
            unsigned long long S = s_S;
            if (p < 3) {
                int sel = -1, dmin = -1;
                for (int d = 255; d >= 0; --d) {
                    if (hcnt[d] == 0u) continue;
                    dmin = d;
                    if (S + hsum[d] > TH9) { sel = d; break; }
                    S += hsum[d];
                }
                if (sel < 0) {                   // whole candidate set kept
                    sel = (dmin >= 0) ? dmin : 0;
                    if (dmin >= 0) S -= hsum[dmin];
                }
                s_prefix = (prefix << 8) | (unsigned)sel;
                s_S = S;
            } else {                             // exact 32-bit granularity
                unsigned tlow = 0u;
                for (int d = 255; d >= 0; --d) {
                    if (hcnt[d] == 0u) continue;
                    if (S <= TH9) { tlow = (unsigned)d; S += hsum[d]; }
                    else break;
                }
                s_tstar = (prefix << 8) | tlow;
            }
        }
        __syncthreads();
    }

    const unsigned tstar = s_tstar;
    unsigned cnt = 0u; unsigned long long sum = 0ull;
    for (int i = t; i < NDIM; i += ROW_T) {
        const float v = srow[i];
        if (__float_as_uint(v) >= tstar) {
            cnt++; sum += (unsigned long long)((double)v * FPSCALE);
        }
    }
    rc[t] = cnt; rs[t] = sum;
    __syncthreads();
    for (int o = ROW_T / 2; o > 0; o >>= 1) {    // fixed-order tree reduce
        if (t < o) { rc[t] += rc[t + o]; rs[t] += rs[t + o]; }
        __syncthreads();
    }
    if (t == 0) {
        thr[r] = tstar;
        const double s = (double)rs[0] * (1.0 / FPSCALE);
        inv_s[r] = (float)(1.0 / s);
        inlier[r] = NDIM - (int)rc[0];
    }
}

// ---------------------------------------------------------------------------
// TDM: DMA one 16x512 f32 tile (row stride 4096) into LDS with hardware
// padding of 2 dwords per 256 dwords -> effective LDS row stride = 516.
// ---------------------------------------------------------------------------
__device__ __forceinline__ unsigned lds_off32(const void* p) {
    return (unsigned)(uintptr_t)p;   // LDS aperture: low 32 bits = LDS address
}

#if HAVE_TDM
__device__ __forceinline__ void tdm_load_chunk(const float* gsrc, unsigned lds_addr) {
    const unsigned long long ga = (unsigned long long)(uintptr_t)gsrc;
    u32x4 g0;
    g0.x = 1u;                                            // count=1, user D#
    g0.y = lds_addr;                                      // LDS dest (bytes)
    g0.z = (unsigned)(ga & 0xFFFFFFFFull);                // global_addr[31:0]
    g0.w = (unsigned)((ga >> 32) & 0x01FFFFFFull)         // global_addr[56:32]
         | 0x80000000u;                                   // type = 2 (image)
    i32x8 g1;
    g1[0] = (int)((2u << 16)        // data_size = 4B
                | (1u << 20)        // pad_enable
                | (7u << 22)        // pad_interval: 256 dwords
                | (1u << 25));      // pad_amount:   2 dwords
    g1[1] = (int)(4096u << 16);                 // tensor_dim0[15:0] @ bits 63:48
    g1[2] = (int)((4096u >> 16) | (16u << 16)); // dim0 hi | tensor_dim1 lo
    g1[3] = (int)(512u << 16);                  // tensor_dim1 hi | tile_dim0
    g1[4] = 16;                                 // tile_dim1 | tile_dim2=0
    g1[5] = 4096;                               // tensor_dim0_stride[31:0]
    g1[6] = 0;                                  // stride0 hi | stride1 lo
    g1[7] = 0;                                  // stride1 hi
    i32x4 g2 = {0, 0, 0, 0};                    // D# group2 (unused, 2D tile)
    i32x4 g3 = {0, 0, 0, 0};                    // D# group3 (unused, 2D tile)
    i32x8 g4 = {0, 0, 0, 0, 0, 0, 0, 0};        // trailing group (zero-filled)
    __builtin_amdgcn_tensor_load_to_lds(g0, g1, g2, g3, g4, 0);
}
#endif

// ---------------------------------------------------------------------------
// Kernel 2: fused mask + renorm + (tgt @ scores_new^T) via V_WMMA_F32_16X16X4
// Block = 16 score-rows; 8 waves split K; TDM double-buffered chunk staging.
// ---------------------------------------------------------------------------
__global__ __launch_bounds__(256) void k_scorr(
    const float* __restrict__ scores, const float* __restrict__ tgt,
    const unsigned* __restrict__ thr, const float* __restrict__ inv_s,
    float* __restrict__ scorr)
{
    __shared__ float sc[2][16 * PADST];          // 2 x 33 KB raw score tiles
    __shared__ float tg16[16 * PADST];           // tgt padded to M=16 (rows 3..15 = 0)
    __shared__ float part[8][3][16];
    __shared__ unsigned s_thr[16];
    __shared__ float s_inv[16];

    const int b = blockIdx.y;
    const int i0 = blockIdx.x * 16;
    const int t = threadIdx.x;
    const int lane = t & 31, w = t >> 5;
    const int m = lane & 15, h = lane >> 4;
    const int padadj = (w >= 4) ? 2 : 0;         // mid-row TDM pad (col >= 256)

    if (t < 16) {
        s_thr[t] = thr[b * NDIM + i0 + t];
        s_inv[t] = inv_s[b * NDIM + i0 + t];
    }
    // zero rows 3..15 of the A tile once (stay zero forever)
    for (int l = t; l < 13 * PADST; l += 256) tg16[3 * PADST + l] = 0.0f;

    const float* sbase = scores + ((size_t)b * NDIM + i0) * NDIM;
    const float* tbase = tgt + (size_t)b * 3 * NDIM;

#if HAVE_TDM
    if (w == 0) {                                // prologue: DMA chunk 0
        tdm_load_chunk(sbase, lds_off32(&sc[0][0]));
        __builtin_amdgcn_s_wait_tensorcnt(0);
    }
#endif
    __syncthreads();
    const unsigned myThr = s_thr[m];

    v8f acc = {};
    for (int ch = 0; ch < NDIM / CHUNK; ++ch) {
        const int c0 = ch * CHUNK;
#if HAVE_TDM
        float* scb = sc[ch & 1];
        if (w == 0 && ch + 1 < NDIM / CHUNK)     // async DMA next chunk
            tdm_load_chunk(sbase + (c0 + CHUNK), lds_off32(&sc[(ch + 1) & 1][0]));
#else
        float* scb = sc[0];
        for (int l = t; l < 16 * (CHUNK / 4); l += 256) {
            const int row = l >> 7;
            const int c4 = (l & 127) << 2;
            const float4 v = *(const float4*)(sbase + (size_t)row * NDIM + c0 + c4);
            float* dst = &scb[row * PADST + c4 + ((c4 >= 256) ? 2 : 0)];
            dst[0] = v.x; dst[1] = v.y; dst[2] = v.z; dst[3] = v.w;
        }
        if (ch + 1 < NDIM / CHUNK)
            __builtin_prefetch(sbase + (size_t)(t >> 4) * NDIM + (c0 + CHUNK) + (t & 15) * 32);
#endif
        // stage tgt rows 0..2 for this chunk (overlaps with DMA)
        for (int l = t; l < 3 * (CHUNK / 4); l += 256) {
            const int row = l >> 7;
            const int c4 = (l & 127) << 2;
            *(float4*)&tg16[row * PADST + c4] =
                *(const float4*)(tbase + (size_t)row * NDIM + c0 + c4);
        }
        __syncthreads();

        // wave w covers chunk columns [w*64, w*64+64) -> 16 wmma steps
        for (int it = 0; it < 16; ++it) {
            const int cc = w * 64 + it * 4 + 2 * h;       // even -> 8B aligned
            v2f bv = *(const v2f*)&scb[m * PADST + cc + padadj];
            v2f av = *(const v2f*)&tg16[m * PADST + cc];
            bv.x = (__float_as_uint(bv.x) >= myThr) ? bv.x : 0.0f;
            bv.y = (__float_as_uint(bv.y) >= myThr) ? bv.y : 0.0f;
            acc = __builtin_amdgcn_wmma_f32_16x16x4_f32(
                false, av, false, bv, (short)0, acc, false, false);
        }
        __syncthreads();
#if HAVE_TDM
        if (w == 0 && ch + 1 < NDIM / CHUNK)
            __builtin_amdgcn_s_wait_tensorcnt(0);        // next buffer resident
        __syncthreads();
#endif
    }

    // lanes 0-15 hold M = 0..7 in acc[0..7]; we need channels M=0..2
    if (h == 0) {
        part[w][0][m] = acc[0];
        part[w][1][m] = acc[1];
        part[w][2][m] = acc[2];
    }
    __syncthreads();
    if (t < 48) {
        const int c = t >> 4, n = t & 15;
        float s = 0.0f;
        for (int ww = 0; ww < 8; ++ww) s += part[ww][c][n];   // fixed order
        s *= s_inv[n];
        scorr[((size_t)b * 3 + c) * NDIM + i0 + n] = s;
    }
}

// ---------------------------------------------------------------------------
// Kernel 3: stable top-k (ascending count, ties by index) + column gather
// ---------------------------------------------------------------------------
__global__ __launch_bounds__(1024) void k_topk_gather(
    const int* __restrict__ inlier, const float* __restrict__ src,
    const float* __restrict__ scorr, float* __restrict__ out, int B)
{
    __shared__ unsigned short cnts[NDIM];        // 8 KB
    __shared__ int sel[K_INL];
    const int b = blockIdx.x;
    const int t = threadIdx.x;

    for (int i = t; i < NDIM; i += 1024)
        cnts[i] = (unsigned short)inlier[b * NDIM + i];
    __syncthreads();

    for (int rr = 0; rr < 4; ++rr) {
        const int i = t * 4 + rr;
        const unsigned ci = cnts[i];
        int rank = 0;
        for (int j = 0; j < NDIM; ++j) {         // LDS broadcast scan
            const unsigned cj = cnts[j];
            rank += (cj < ci) || (cj == ci && j < i);
        }
        if (rank < K_INL) sel[rank] = i;
    }
    __syncthreads();

    const int idx = sel[t];
    const size_t half = (size_t)B * 3 * K_INL;   // src_new block size
    for (int c = 0; c < 3; ++c) {
        out[((size_t)b * 3 + c) * K_INL + t] =
            src[((size_t)b * 3 + c) * NDIM + idx];
        out[half + ((size_t)b * 3 + c) * K_INL + t] =
            scorr[((size_t)b * 3 + c) * NDIM + idx];
    }
}

// ---------------------------------------------------------------------------
extern "C" void kernel_launch(void* const* d_in, const int* in_sizes, int n_in,
                              void* d_out, int out_size, void* d_ws, size_t ws_size,
                              hipStream_t stream) {
    const float* src    = (const float*)d_in[0];   // [B,3,N]
    const float* tgt    = (const float*)d_in[1];   // [B,3,N]
    const float* scores = (const float*)d_in[2];   // [B,N,N]
    float* out = (float*)d_out;

    const int B = in_sizes[0] / (3 * NDIM);

    // workspace layout
    char* ws = (char*)d_ws;
    unsigned* thr   = (unsigned*)ws;                       ws += (size_t)B * NDIM * sizeof(unsigned);
    float*    invs  = (float*)ws;                          ws += (size_t)B * NDIM * sizeof(float);
    int*      inl   = (int*)ws;                            ws += (size_t)B * NDIM * sizeof(int);
    float*    scorr = (float*)ws;                          // B*3*N floats

    k_threshold<<<dim3(B * NDIM), dim3(ROW_T), 0, stream>>>(scores, thr, invs, inl);
    k_scorr<<<dim3(NDIM / 16, B), dim3(256), 0, stream>>>(scores, tgt, thr, invs, scorr);
    k_topk_gather<<<dim3(B), dim3(1024), 0, stream>>>(inl, src, scorr, out, B);
}